// GCN_v3_5652176962025
// MI455X (gfx1250) — compile-verified
//
#include <hip/hip_runtime.h>

#define HDIM 128
#define EPSV 1e-5f

typedef float v2f __attribute__((ext_vector_type(2)));
typedef float v8f __attribute__((ext_vector_type(8)));

static inline int cdiv(long long a, long long b) { return (int)((a + b - 1) / b); }

// ---------------- degree / normalization ----------------
__global__ __launch_bounds__(256) void k_deg(const int* __restrict__ dst,
                                             float* __restrict__ deg, int E) {
  int i = blockIdx.x * 256 + threadIdx.x;
  if (i < E) atomicAdd(&deg[dst[i]], 1.0f);
}

__global__ __launch_bounds__(256) void k_dinv(float* __restrict__ deg, int N) {
  int i = blockIdx.x * 256 + threadIdx.x;
  if (i < N) deg[i] = rsqrtf(deg[i] + 1.0f);
}

// ---------------- fp32 WMMA GEMM: C[nrows x 128] = A[nrows x 128] * W[128 x 128] ----------------
// One 16x16 output tile per wave; 8 waves/block cover 16 rows x 128 cols.
// A frag (16x4 f32): lane l -> M = l&15, VGPR0/1 hold K = (l>>4)*2, (l>>4)*2+1
// B frag (4x16 f32): lane l -> N = l&15, VGPR0/1 hold K = (l>>4)*2, (l>>4)*2+1
// C/D (16x16 f32):   VGPR r, lane l -> M = r + (l>>4)*8, N = l&15
__global__ __launch_bounds__(256) void k_gemm(const float* __restrict__ A,
                                              const float* __restrict__ W,
                                              float* __restrict__ C, int nrows) {
  const int wave = threadIdx.x >> 5;
  const int lane = threadIdx.x & 31;
  const int m    = lane & 15;
  const int half = lane >> 4;
  const int row0 = blockIdx.x * 16;
  const int col0 = wave * 16;
  const int row  = row0 + m;
  const bool rowok = row < nrows;
  const float* aptr = A + (size_t)(rowok ? row : 0) * HDIM;
  const int kb = half * 2;
  v8f acc = {0.f, 0.f, 0.f, 0.f, 0.f, 0.f, 0.f, 0.f};
#pragma unroll
  for (int k = 0; k < HDIM; k += 4) {
    v2f a, b;
    a.x = aptr[k + kb];
    a.y = aptr[k + kb + 1];
    a.x = rowok ? a.x : 0.f;   // v_cndmask, no EXEC divergence before WMMA
    a.y = rowok ? a.y : 0.f;
    b.x = W[(size_t)(k + kb) * HDIM + col0 + m];
    b.y = W[(size_t)(k + kb + 1) * HDIM + col0 + m];
    acc = __builtin_amdgcn_wmma_f32_16x16x4_f32(false, a, false, b, (short)0, acc,
                                                false, false);
  }
#pragma unroll
  for (int r = 0; r < 8; ++r) {
    int orow = row0 + r + half * 8;
    if (orow < nrows) C[(size_t)orow * HDIM + col0 + m] = acc[r];
  }
}

// ---------------- edge scatter: agg[dst] += h[src] * dinv[src]*dinv[dst] ----------------
// One wave per edge: 32 lanes x float4 = 128 features, coalesced gather + atomic scatter (L2-resident).
__global__ __launch_bounds__(256) void k_scatter(const float* __restrict__ h,
                                                 const int* __restrict__ src,
                                                 const int* __restrict__ dst,
                                                 const float* __restrict__ dinv,
                                                 float* __restrict__ agg, int E) {
  long long tid = (long long)blockIdx.x * 256 + threadIdx.x;
  int e = (int)(tid >> 5);
  if (e >= E) return;
  int f4 = ((int)tid & 31) * 4;
  int s = src[e], d = dst[e];
  float nrm = dinv[s] * dinv[d];
  const float4 v = *(const float4*)(h + (size_t)s * HDIM + f4);
  float* ap = agg + (size_t)d * HDIM + f4;
  atomicAdd(ap + 0, v.x * nrm);
  atomicAdd(ap + 1, v.y * nrm);
  atomicAdd(ap + 2, v.z * nrm);
  atomicAdd(ap + 3, v.w * nrm);
}

// ---------------- self-loop + bias, fused BN statistics ----------------
// 128 threads (one channel each) x 256 rows per block; register accumulation, 2 atomics/channel/block.
#define ROWS_PER_BLK 256
__global__ __launch_bounds__(128) void k_selfloop_stats(const float* __restrict__ h,
                                                        const float* __restrict__ dinv,
                                                        const float* __restrict__ bias,
                                                        float* __restrict__ y,
                                                        float* __restrict__ sums, int N) {
  int c  = threadIdx.x;
  int r0 = blockIdx.x * ROWS_PER_BLK;
  float b = bias[c];
  float s = 0.f, s2 = 0.f;
  for (int i = 0; i < ROWS_PER_BLK; ++i) {
    int r = r0 + i;
    if (r >= N) break;
    float di = dinv[r];
    size_t idx = (size_t)r * HDIM + c;
    float v = y[idx] + h[idx] * di * di + b;
    y[idx] = v;
    s  += v;
    s2 += v * v;
  }
  atomicAdd(&sums[c], s);
  atomicAdd(&sums[HDIM + c], s2);
}

// ---------------- BN scale/shift from stats ----------------
__global__ __launch_bounds__(128) void k_bn_params(const float* __restrict__ sums,
                                                   const float* __restrict__ g,
                                                   const float* __restrict__ be,
                                                   float* __restrict__ bnp, float invN) {
  int c = threadIdx.x;
  float mean = sums[c] * invN;
  float var  = sums[HDIM + c] * invN - mean * mean;
  float sc   = g[c] * rsqrtf(var + EPSV);
  bnp[c]        = sc;
  bnp[HDIM + c] = be[c] - mean * sc;
}

// ---------------- apply BN + ReLU (+ optional residual) ----------------
__global__ __launch_bounds__(256) void k_bn_apply(const float* __restrict__ y,
                                                  const float* __restrict__ bnp,
                                                  const float* __restrict__ res,
                                                  float* __restrict__ out, int n4,
                                                  int has_res) {
  int i = blockIdx.x * 256 + threadIdx.x;
  if (i >= n4) return;
  int c4 = (i & 31) * 4;  // HDIM/4 == 32 float4 per row
  float4 v  = ((const float4*)y)[i];
  float4 sc = *(const float4*)(bnp + c4);
  float4 sh = *(const float4*)(bnp + HDIM + c4);
  float4 r;
  r.x = fmaxf(v.x * sc.x + sh.x, 0.f);
  r.y = fmaxf(v.y * sc.y + sh.y, 0.f);
  r.z = fmaxf(v.z * sc.z + sh.z, 0.f);
  r.w = fmaxf(v.w * sc.w + sh.w, 0.f);
  if (has_res) {
    float4 rv = ((const float4*)res)[i];
    r.x += rv.x; r.y += rv.y; r.z += rv.z; r.w += rv.w;
  }
  ((float4*)out)[i] = r;
}

// ---------------- pooling: exploit sorted batch, flush atomics on graph change ----------------
__global__ __launch_bounds__(128) void k_pool(const float* __restrict__ h,
                                              const int* __restrict__ batch,
                                              float* __restrict__ xsum,
                                              float* __restrict__ cnt, int N) {
  int c  = threadIdx.x;
  int r0 = blockIdx.x * 256;
  int rend = min(r0 + 256, N);
  if (r0 >= rend) return;
  int curg = batch[r0];
  float acc = 0.f, cacc = 0.f;
  for (int r = r0; r < rend; ++r) {
    int g = batch[r];
    if (g != curg) {
      atomicAdd(&xsum[(size_t)curg * HDIM + c], acc);
      if (c == 0) atomicAdd(&cnt[curg], cacc);
      curg = g; acc = 0.f; cacc = 0.f;
    }
    acc  += h[(size_t)r * HDIM + c];
    cacc += 1.f;
  }
  atomicAdd(&xsum[(size_t)curg * HDIM + c], acc);
  if (c == 0) atomicAdd(&cnt[curg], cacc);
}

// ---------------- MLP head: [G x 2H] -> ReLU(H) -> 1 ----------------
__global__ __launch_bounds__(128) void k_mlp(const float* __restrict__ xsum,
                                             const float* __restrict__ cnt,
                                             const float* __restrict__ Wm1,
                                             const float* __restrict__ bm1,
                                             const float* __restrict__ Wm2,
                                             const float* __restrict__ bm2,
                                             float* __restrict__ out) {
  int g = blockIdx.x;
  int j = threadIdx.x;
  float inv = 1.f / fmaxf(cnt[g], 1.f);
  float acc = bm1[j];
  for (int k = 0; k < HDIM; ++k) {
    float xs = xsum[(size_t)g * HDIM + k];
    acc += xs * Wm1[(size_t)k * HDIM + j];                 // sum half of concat
    acc += (xs * inv) * Wm1[(size_t)(HDIM + k) * HDIM + j]; // mean half of concat
  }
  float hv = fmaxf(acc, 0.f) * Wm2[j];
  __shared__ float red[128];
  red[j] = hv;
  __syncthreads();
  for (int s = 64; s > 0; s >>= 1) {
    if (j < s) red[j] += red[j + s];
    __syncthreads();
  }
  if (j == 0) out[g] = red[0] + bm2[0];
}

// ---------------- driver ----------------
extern "C" void kernel_launch(void* const* d_in, const int* in_sizes, int n_in,
                              void* d_out, int out_size, void* d_ws, size_t ws_size,
                              hipStream_t stream) {
  const float* x    = (const float*)d_in[0];
  const int* eidx   = (const int*)d_in[1];
  const int* batch  = (const int*)d_in[2];
  const float* W1   = (const float*)d_in[3];
  const float* b1   = (const float*)d_in[4];
  const float* W2   = (const float*)d_in[5];
  const float* b2   = (const float*)d_in[6];
  const float* W3   = (const float*)d_in[7];
  const float* b3   = (const float*)d_in[8];
  const float* g1   = (const float*)d_in[9];
  const float* be1  = (const float*)d_in[10];
  const float* g2   = (const float*)d_in[11];
  const float* be2  = (const float*)d_in[12];
  const float* g3   = (const float*)d_in[13];
  const float* be3  = (const float*)d_in[14];
  const float* Wm1  = (const float*)d_in[15];
  const float* bm1  = (const float*)d_in[16];
  const float* Wm2  = (const float*)d_in[17];
  const float* bm2  = (const float*)d_in[18];
  float* out = (float*)d_out;

  const int N = in_sizes[0] / HDIM;
  const int E = in_sizes[1] / 2;
  const int G = out_size;
  const int* src = eidx;
  const int* dst = eidx + E;

  // workspace layout (floats)
  float* B0   = (float*)d_ws;
  float* B1   = B0 + (size_t)N * HDIM;
  float* B2   = B1 + (size_t)N * HDIM;
  float* dinv = B2 + (size_t)N * HDIM;
  float* sums = dinv + N;            // 2*HDIM
  float* bnp  = sums + 2 * HDIM;     // 2*HDIM (scale, shift)
  float* xsum = bnp + 2 * HDIM;      // G*HDIM
  float* cnt  = xsum + (size_t)G * HDIM; // G
  (void)n_in; (void)ws_size;

  const size_t featBytes = (size_t)N * HDIM * sizeof(float);
  const int gGemm    = cdiv(N, 16);
  const int gScatter = cdiv((long long)E * 32, 256);
  const int gRows    = cdiv(N, ROWS_PER_BLK);
  const int gElem4   = cdiv((long long)N * HDIM / 4, 256);
  const float invN   = 1.0f / (float)N;

  // degree -> dinv
  hipMemsetAsync(dinv, 0, (size_t)N * sizeof(float), stream);
  k_deg<<<cdiv(E, 256), 256, 0, stream>>>(dst, dinv, E);
  k_dinv<<<cdiv(N, 256), 256, 0, stream>>>(dinv, N);

  // ---- layer 1: in = x, out -> B2 (no residual)
  k_gemm<<<gGemm, 256, 0, stream>>>(x, W1, B0, N);
  hipMemsetAsync(B1, 0, featBytes, stream);
  k_scatter<<<gScatter, 256, 0, stream>>>(B0, src, dst, dinv, B1, E);
  hipMemsetAsync(sums, 0, 2 * HDIM * sizeof(float), stream);
  k_selfloop_stats<<<gRows, 128, 0, stream>>>(B0, dinv, b1, B1, sums, N);
  k_bn_params<<<1, 128, 0, stream>>>(sums, g1, be1, bnp, invN);
  k_bn_apply<<<gElem4, 256, 0, stream>>>(B1, bnp, nullptr, B2, N * HDIM / 4, 0);

  // ---- layer 2: in = B2, out -> B0 (residual = B2)
  k_gemm<<<gGemm, 256, 0, stream>>>(B2, W2, B0, N);
  hipMemsetAsync(B1, 0, featBytes, stream);
  k_scatter<<<gScatter, 256, 0, stream>>>(B0, src, dst, dinv, B1, E);
  hipMemsetAsync(sums, 0, 2 * HDIM * sizeof(float), stream);
  k_selfloop_stats<<<gRows, 128, 0, stream>>>(B0, dinv, b2, B1, sums, N);
  k_bn_params<<<1, 128, 0, stream>>>(sums, g2, be2, bnp, invN);
  k_bn_apply<<<gElem4, 256, 0, stream>>>(B1, bnp, B2, B0, N * HDIM / 4, 1);

  // ---- layer 3: in = B0, out -> B2 (residual = B0)
  k_gemm<<<gGemm, 256, 0, stream>>>(B0, W3, B2, N);
  hipMemsetAsync(B1, 0, featBytes, stream);
  k_scatter<<<gScatter, 256, 0, stream>>>(B2, src, dst, dinv, B1, E);
  hipMemsetAsync(sums, 0, 2 * HDIM * sizeof(float), stream);
  k_selfloop_stats<<<gRows, 128, 0, stream>>>(B2, dinv, b3, B1, sums, N);
  k_bn_params<<<1, 128, 0, stream>>>(sums, g3, be3, bnp, invN);
  k_bn_apply<<<gElem4, 256, 0, stream>>>(B1, bnp, B0, B2, N * HDIM / 4, 1);

  // ---- pooling + MLP head
  hipMemsetAsync(xsum, 0, ((size_t)G * HDIM + G) * sizeof(float), stream);
  k_pool<<<cdiv(N, 256), 128, 0, stream>>>(B2, batch, xsum, cnt, N);
  k_mlp<<<G, 128, 0, stream>>>(xsum, cnt, Wm1, bm1, Wm2, bm2, out);
}